// YOLOLoss_78280073937323
// MI455X (gfx1250) — compile-verified
//
#include <hip/hip_runtime.h>
#include <math.h>

// ---------------- problem constants (from reference setup_inputs) ------------
#define BS      8
#define TNUM    50
#define HH      76
#define WW      76
#define AA      3
#define CC      80
#define CH_TOT  255          // 3 * (5 + 80)
#define HW      (HH * WW)    // 5776
#define CELLS_PB (AA * HW)   // 17328 cells per batch image
#define NTOT    (BS * CELLS_PB)

#define IGNORE_THRESH 0.5f
#define PI_F 3.14159265358979323846f

// scaled anchors = ANCHORS / STRIDE (STRIDE = 8)
__constant__ float c_sa[9][2] = {
    {17.75f, 13.75f}, {24.0f, 30.375f}, {57.375f, 50.125f},
    {4.5f,   9.375f}, {9.5f,  6.875f},  {9.0f,   18.25f},
    {1.5f,   2.0f},   {2.375f, 4.5f},   {5.0f,   3.5f}};
// layer anchors = sa[6..8]
__constant__ float c_law[3] = {1.5f, 2.375f, 5.0f};
__constant__ float c_lah[3] = {2.0f, 4.5f, 3.5f};

typedef float v2f __attribute__((ext_vector_type(2)));
typedef float v8f __attribute__((ext_vector_type(8)));

__device__ __forceinline__ float sigmoidf_(float x) {
    return 1.0f / (1.0f + expf(-x));
}
__device__ __forceinline__ float bcef_(float p, float t) {
    const float eps = 1e-7f;
    p = fminf(fmaxf(p, eps), 1.0f - eps);
    return -t * logf(p) - (1.0f - t) * logf(1.0f - p);
}

// ---------------- kernel 0: zero mask + accumulators -------------------------
__global__ void k_zero(float* acc, float* mask) {
    int i = blockIdx.x * blockDim.x + threadIdx.x;
    if (i < 8) acc[i] = 0.0f;
    if (i < NTOT) mask[i] = 0.0f;
}

// ---------------- kernel 1: scatter targets into the grid --------------------
__global__ void k_scatter(const float* __restrict__ tb, const int* __restrict__ tc,
                          float* mask, float* tbox, float* bscale, int* clsidx) {
    int idx = blockIdx.x * blockDim.x + threadIdx.x;
    if (idx >= BS * TNUM) return;
    int b = idx / TNUM;
    const float* p = tb + (size_t)idx * 4;
    float gx = p[0] * WW, gy = p[1] * HH, gw = p[2] * WW, gh = p[3] * HH;
    // argmax IoU vs 9 anchors (first max wins, matching jnp.argmax)
    int best = 0; float bestr = -1e30f;
    #pragma unroll
    for (int k = 0; k < 9; ++k) {
        float aw = c_sa[k][0], ah = c_sa[k][1];
        float inter = fminf(gw, aw) * fminf(gh, ah);
        float uni = gw * gh + aw * ah - inter;
        float r = inter / uni;
        if (r > bestr) { bestr = r; best = k; }
    }
    int gi = (int)floorf(gx), gj = (int)floorf(gy);
    bool valid = (best >= 6) && (best < 9) && (gi < WW) && (gj < HH);
    if (!valid) return;
    int a = best - 6;
    int lin = ((b * AA + a) * HH + gj) * WW + gi;
    mask[lin] = 1.0f;
    tbox[4 * lin + 0] = gx;
    tbox[4 * lin + 1] = gy;
    tbox[4 * lin + 2] = gw;
    tbox[4 * lin + 3] = gh;
    bscale[lin] = p[2] * p[3];          // normalized w*h (for 2 - wh scale)
    clsidx[lin] = tc[idx];
}

// ---------------- kernel 2: fused decode + IoU-max + losses ------------------
// grid = (ceil(CELLS_PB/256), BS); block = 256.  Reads only 5/85 channels for
// every cell; the 80 class channels are touched only at <=400 masked cells.
__global__ __launch_bounds__(256) void k_main(
    const float* __restrict__ in, const float* __restrict__ tb,
    const float* __restrict__ mask, const float* __restrict__ tbox,
    const float* __restrict__ bscale, const int* __restrict__ clsidx,
    float* acc) {
    __shared__ float sg1x[TNUM], sg1y[TNUM], sg2x[TNUM], sg2y[TNUM], sga[TNUM];
    __shared__ float spc[256], spl[256];

    const int b = blockIdx.y;
    const int tid = threadIdx.x;

    // stage this batch image's 50 GT boxes (corners + area) into LDS
    if (tid < TNUM) {
        const float* p = tb + ((size_t)b * TNUM + tid) * 4;
        float gx = p[0] * WW, gy = p[1] * HH, gw = p[2] * WW, gh = p[3] * HH;
        sg1x[tid] = gx - 0.5f * gw; sg2x[tid] = gx + 0.5f * gw;
        sg1y[tid] = gy - 0.5f * gh; sg2y[tid] = gy + 0.5f * gh;
        sga[tid]  = gw * gh;
    }
    __syncthreads();

    const int cell = blockIdx.x * blockDim.x + tid;
    float pc = 0.0f, pl = 0.0f;

    if (cell < CELLS_PB) {
        const int a   = cell / HW;
        const int rem = cell % HW;
        const int i   = rem / WW;   // h (row) index -> grid_y
        const int j   = rem % WW;   // w (col) index -> grid_x
        const size_t base = (((size_t)b * CH_TOT + a * (5 + CC)) * HH + i) * WW + j;

        const float t0 = in[base];
        const float t1 = in[base + 1 * (size_t)HW];
        const float t2 = in[base + 2 * (size_t)HW];
        const float t3 = in[base + 3 * (size_t)HW];
        const float t4 = in[base + 4 * (size_t)HW];

        const float px = sigmoidf_(t0) + (float)j;
        const float py = sigmoidf_(t1) + (float)i;
        const float pw = expf(t2) * c_law[a];
        const float ph = expf(t3) * c_lah[a];
        const float conf = sigmoidf_(t4);

        const float p1x = px - 0.5f * pw, p2x = px + 0.5f * pw;
        const float p1y = py - 0.5f * ph, p2y = py + 0.5f * ph;
        const float pa  = pw * ph;

        // max IoU over the 50 GT boxes (no eps in denominator, per reference)
        float ioumax = -1e30f;
        #pragma unroll 5
        for (int t = 0; t < TNUM; ++t) {
            float iw = fmaxf(fminf(sg2x[t], p2x) - fmaxf(sg1x[t], p1x), 0.0f);
            float ih = fmaxf(fminf(sg2y[t], p2y) - fmaxf(sg1y[t], p1y), 0.0f);
            float inter = iw * ih;
            float iou = inter / (sga[t] + pa - inter);
            ioumax = fmaxf(ioumax, iou);
        }

        const int lin = b * CELLS_PB + cell;
        const float m = mask[lin];
        float noobj = 1.0f - m;
        if (ioumax > IGNORE_THRESH) noobj = 0.0f;

        pc = bcef_(conf, m) * (m + noobj);          // conf loss partial

        if (m > 0.0f) {
            // ----- CIoU vs scattered target box -----
            const float gx = tbox[4 * lin + 0], gy = tbox[4 * lin + 1];
            const float gw = tbox[4 * lin + 2], gh = tbox[4 * lin + 3];
            const float b2x1 = gx - 0.5f * gw, b2x2 = gx + 0.5f * gw;
            const float b2y1 = gy - 0.5f * gh, b2y2 = gy + 0.5f * gh;
            float iw = fmaxf(fminf(p2x, b2x2) - fmaxf(p1x, b2x1), 0.0f);
            float ih = fmaxf(fminf(p2y, b2y2) - fmaxf(p1y, b2y1), 0.0f);
            float inter = iw * ih;
            float uni = pw * ph + gw * gh - inter;
            float iou = inter / fmaxf(uni, 1e-6f);
            float cd = (px - gx) * (px - gx) + (py - gy) * (py - gy);
            float ew = fmaxf(fmaxf(p2x, b2x2) - fminf(p1x, b2x1), 0.0f);
            float eh = fmaxf(fmaxf(p2y, b2y2) - fminf(p1y, b2y1), 0.0f);
            float ed = ew * ew + eh * eh;
            float ciou = iou - cd / fmaxf(ed, 1e-6f);
            float dv = atanf(pw / fmaxf(ph, 1e-6f)) - atanf(gw / fmaxf(gh, 1e-6f));
            float v = (4.0f / (PI_F * PI_F)) * dv * dv;
            float alpha = v / fmaxf(1.0f - iou + v, 1e-6f);
            ciou = ciou - alpha * v;
            pl = (1.0f - ciou) * (2.0f - bscale[lin]); // loc loss partial

            // ----- class BCE: only masked cells touch the 80 class channels --
            const int ci = clsidx[lin];
            float cl = 0.0f;
            const size_t cbase = base + 5 * (size_t)HW;
            #pragma unroll 8
            for (int c = 0; c < CC; ++c) {
                float s = sigmoidf_(in[cbase + (size_t)c * HW]);
                cl += bcef_(s, (c == ci) ? 1.0f : 0.0f);
            }
            atomicAdd(&acc[1], cl);   // rare (<=400 threads total hit this)
        }
    }

    // -------- block reduction of (pc, pl) through the WMMA pipe --------------
    // D = A(16x4) * ones(4x16) + C  => D[m][n] = rowsum(A[m]); with B = ones,
    // one V_WMMA_F32_16X16X4_F32 folds 64 fp32 partials into 16 row sums.
    spc[tid] = pc;
    spl[tid] = pl;
    __syncthreads();

    if (tid < 32) {   // wave 0 only -> EXEC all-ones inside the wave
        v2f ones; ones.x = 1.0f; ones.y = 1.0f;
        v8f accc = {0.f, 0.f, 0.f, 0.f, 0.f, 0.f, 0.f, 0.f};
        v8f accl = {0.f, 0.f, 0.f, 0.f, 0.f, 0.f, 0.f, 0.f};
        #pragma unroll
        for (int t = 0; t < 4; ++t) {
            v2f av; av.x = spc[t * 64 + tid]; av.y = spc[t * 64 + 32 + tid];
            accc = __builtin_amdgcn_wmma_f32_16x16x4_f32(
                false, av, false, ones, (short)0, accc, false, false);
            v2f bv; bv.x = spl[t * 64 + tid]; bv.y = spl[t * 64 + 32 + tid];
            accl = __builtin_amdgcn_wmma_f32_16x16x4_f32(
                false, bv, false, ones, (short)0, accl, false, false);
        }
        // each lane now holds 8 row sums (rows 0-7 in lanes 0-15, rows 8-15 in
        // lanes 16-31, replicated over n) -> vector add + xor-16 finishes it.
        float sc = accc[0] + accc[1] + accc[2] + accc[3] +
                   accc[4] + accc[5] + accc[6] + accc[7];
        float sl = accl[0] + accl[1] + accl[2] + accl[3] +
                   accl[4] + accl[5] + accl[6] + accl[7];
        sc += __shfl_xor(sc, 16, 32);
        sl += __shfl_xor(sl, 16, 32);
        if (tid == 0) {
            atomicAdd(&acc[0], sc);
            atomicAdd(&acc[2], sl);
        }
    }
}

// ---------------- kernel 3: finalize the 4-scalar tuple ----------------------
__global__ void k_finalize(const float* __restrict__ acc, float* __restrict__ out) {
    if (blockIdx.x == 0 && threadIdx.x == 0) {
        float lconf = acc[0] / (float)BS;
        float lcls  = acc[1] / (float)BS;
        float lloc  = acc[2] / (float)BS;
        out[0] = lconf + lcls + lloc;
        out[1] = lconf;
        out[2] = lcls;
        out[3] = lloc;
    }
}

// ---------------- launcher ---------------------------------------------------
extern "C" void kernel_launch(void* const* d_in, const int* in_sizes, int n_in,
                              void* d_out, int out_size, void* d_ws, size_t ws_size,
                              hipStream_t stream) {
    const float* input = (const float*)d_in[0];          // (8, 255, 76, 76) f32
    const float* tboxes = (const float*)d_in[1];         // (8, 50, 4) f32
    const int*   tcls   = (const int*)d_in[2];           // (8, 50) i32
    float* out = (float*)d_out;                          // 4 scalars

    // workspace partition (floats): acc[8] | mask[N] | tbox[4N] | bscale[N] | cls[N]
    float* ws     = (float*)d_ws;
    float* acc    = ws;
    float* mask   = ws + 8;
    float* tbox   = mask + NTOT;
    float* bscale = tbox + 4 * (size_t)NTOT;
    int*   clsidx = (int*)(bscale + NTOT);

    (void)in_sizes; (void)n_in; (void)out_size; (void)ws_size;

    k_zero<<<(NTOT + 255) / 256, 256, 0, stream>>>(acc, mask);
    k_scatter<<<(BS * TNUM + 255) / 256, 256, 0, stream>>>(
        tboxes, tcls, mask, tbox, bscale, clsidx);
    dim3 grid((CELLS_PB + 255) / 256, BS);
    k_main<<<grid, 256, 0, stream>>>(input, tboxes, mask, tbox, bscale, clsidx, acc);
    k_finalize<<<1, 64, 0, stream>>>(acc, out);
}